// FlattenedWindowMapping_10505490006554
// MI455X (gfx1250) — compile-verified
//
#include <hip/hip_runtime.h>
#include <stdint.h>

// ---------------- problem constants (from reference) ----------------
#define GRP   69      // GROUP_SIZE
#define WWX   12      // WINDOW_SHAPE x
#define WWY   12      // WINDOW_SHAPE y
#define WWZ   32      // WINDOW_SHAPE z
#define SHX   (WWX/2) // SHIFT=True
#define SHY   (WWY/2)
#define SHZ   (WWZ/2)
#define MAXB  16
#define TPB   256     // threads per block (8 wave32s)
#define ITEMS 4       // items per thread in radix tiles
#define TILE  (TPB*ITEMS)
#define SCB   256     // blocks used by the parallel hist scan

// meta layout (int32 units inside workspace):
//  [0..15] counts, [16..32] bs (17), [33..48] counts_p, [49..65] bsp (17), [66] B
#define M_COUNTS 0
#define M_BS     16
#define M_CP     33
#define M_BSP    49
#define M_B      66
#define META_INTS 128

// ---------------- tiny kernels ----------------
__global__ void fwm_zero_meta(int* meta) {
    if (threadIdx.x < META_INTS) meta[threadIdx.x] = 0;
}

__global__ void fwm_count(const int* __restrict__ coords, int* __restrict__ meta, int N) {
    __shared__ int h[MAXB];
    if (threadIdx.x < MAXB) h[threadIdx.x] = 0;
    __syncthreads();
    int base = blockIdx.x * TILE;
    for (int r = 0; r < ITEMS; ++r) {
        int i = base + r * TPB + threadIdx.x;
        if (i < N) {
            if (i + TILE < N) __builtin_prefetch(&coords[4 * (i + TILE)], 0, 1); // global_prefetch_b8
            int b = coords[4 * i];
            if (b >= 0 && b < MAXB) atomicAdd(&h[b], 1);
        }
    }
    __syncthreads();
    if (threadIdx.x < MAXB) {
        int v = h[threadIdx.x];
        if (v) atomicAdd(&meta[M_COUNTS + threadIdx.x], v);
    }
}

__global__ void fwm_scan_meta(int* meta, const int* bsz_ptr) {
    if (threadIdx.x != 0) return;
    int B = *bsz_ptr;
    if (B < 1) B = 1;
    if (B > MAXB) B = MAXB;
    meta[M_BS] = 0;
    meta[M_BSP] = 0;
    for (int b = 0; b < B; ++b) {
        int c = meta[M_COUNTS + b];
        int cp = ((c + GRP - 1) / GRP) * GRP;
        meta[M_CP + b] = cp;
        meta[M_BS + b + 1]  = meta[M_BS + b]  + c;
        meta[M_BSP + b + 1] = meta[M_BSP + b] + cp;
    }
    meta[M_B] = B;
}

// ---------------- keys (+win2flat) with async global->LDS staging ----------------
__global__ void fwm_keys(const int* __restrict__ coords, const int* __restrict__ meta,
                         const int* __restrict__ sshape, int N,
                         int2* __restrict__ pairOut, int* __restrict__ win2flat, int axis) {
    __shared__ int4 tile[TPB];
    int i = blockIdx.x * TPB + threadIdx.x;

    if (i < N) {
        uint32_t ldsoff = (uint32_t)(uintptr_t)(&tile[threadIdx.x]);
        const int4* gp = ((const int4*)coords) + i;
        asm volatile("global_load_async_to_lds_b128 %0, %1, off"
                     :: "v"(ldsoff), "v"(gp) : "memory");
    }
    asm volatile("s_wait_asynccnt 0" ::: "memory");
    __syncthreads();
    if (i >= N) return;

    int4 c = tile[threadIdx.x];            // (b, z, y, x)
    int b = c.x;
    int z = c.y + SHZ;
    int y = c.z + SHY;
    int x = c.w + SHX;

    int sz = sshape[0], sy = sshape[1], sx = sshape[2];
    int mwx = (sx + WWX - 1) / WWX + 1;
    int mwy = (sy + WWY - 1) / WWY + 1;
    int mwz = (sz + WWZ - 1) / WWZ + 1;
    int mps = mwx * mwy * mwz;
    const int vol = WWX * WWY * WWZ;

    int wx = x / WWX, cx = x % WWX;
    int wy = y / WWY, cy = y % WWY;
    int wz = z / WWZ, cz = z % WWZ;

    int key;
    if (axis == 0) {
        int bwi = b * mps + wx * (mwy * mwz) + wy * mwz + wz;
        key = bwi * vol + cx * (WWY * WWZ) + cy * WWZ + cz;
    } else {
        int bwi = b * mps + wy * (mwx * mwz) + wx * mwz + wz;
        key = bwi * vol + cy * (WWX * WWZ) + cx * WWZ + cz;
    }
    pairOut[i] = make_int2(key, i);

    if (win2flat) {
        int off = meta[M_BSP + b] - meta[M_BS + b];
        win2flat[i] = i + off;
    }
}

// ---------------- flat2win ----------------
__global__ void fwm_flat2win(const int* __restrict__ meta, int Np, int* __restrict__ out) {
    __shared__ int m[META_INTS];
    if (threadIdx.x < META_INTS) m[threadIdx.x] = meta[threadIdx.x];
    __syncthreads();
    int j = blockIdx.x * blockDim.x + threadIdx.x;
    if (j >= Np) return;
    int B = m[M_B];
    int b = 0;
    while (b + 1 < B && m[M_BSP + b + 1] <= j) ++b;
    int bs   = m[M_BS + b];
    int bsp  = m[M_BSP + b];
    int bsp1 = m[M_BSP + b + 1];
    int num  = m[M_COUNTS + b];
    int nump = m[M_CP + b];
    int off  = bsp - bs;
    int r    = num % GRP;
    bool has_pad = (num != nump);
    int tail_start = has_pad ? (bsp1 - GRP + r) : bsp1;
    int v;
    if (j < tail_start) {
        v = j - off;
    } else if (nump != GRP) {
        v = j - GRP - off;
    } else {
        int t = j - tail_start;
        int mm = (num > 0) ? num : 1;
        v = bs + t % mm;
    }
    out[j] = v;
}

// ---------------- radix sort: per-tile histogram (digit-major output) ----------------
__global__ void fwm_hist(const int2* __restrict__ pairIn, int N, int shift, int NB,
                         int* __restrict__ hist) {
    __shared__ int h[256];
    h[threadIdx.x] = 0;
    __syncthreads();
    int base = blockIdx.x * TILE;
    for (int r = 0; r < ITEMS; ++r) {
        int i = base + r * TPB + threadIdx.x;
        if (i < N) {
            unsigned d = ((unsigned)pairIn[i].x >> shift) & 255u;
            atomicAdd(&h[d], 1);
        }
    }
    __syncthreads();
    hist[threadIdx.x * NB + blockIdx.x] = h[threadIdx.x];
}

// ---------------- parallel exclusive scan over hist (3 kernels) ----------------
__global__ void fwm_scan1(const int* __restrict__ a, int M, int* __restrict__ partial) {
    __shared__ int s[TPB];
    int chunk = (M + SCB - 1) / SCB;
    int lo = blockIdx.x * chunk;
    int hi = lo + chunk; if (hi > M) hi = M;
    int L = (chunk + TPB - 1) / TPB;
    int tlo = lo + threadIdx.x * L;
    int thi = tlo + L; if (thi > hi) thi = hi;
    int acc = 0;
    for (int k = tlo; k < thi; ++k) acc += a[k];
    s[threadIdx.x] = acc;
    __syncthreads();
    if (threadIdx.x == 0) {
        int tot = 0;
        for (int t = 0; t < TPB; ++t) tot += s[t];
        partial[blockIdx.x] = tot;
    }
}

__global__ void fwm_scan2(int* partial) {
    if (threadIdx.x == 0) {
        int run = 0;
        for (int b = 0; b < SCB; ++b) { int tmp = partial[b]; partial[b] = run; run += tmp; }
    }
}

__global__ void fwm_scan3(int* __restrict__ a, int M, const int* __restrict__ partial) {
    __shared__ int s[TPB];
    int chunk = (M + SCB - 1) / SCB;
    int lo = blockIdx.x * chunk;
    int hi = lo + chunk; if (hi > M) hi = M;
    int L = (chunk + TPB - 1) / TPB;
    int tlo = lo + threadIdx.x * L;
    int thi = tlo + L; if (thi > hi) thi = hi;
    int acc = 0;
    for (int k = tlo; k < thi; ++k) acc += a[k];
    s[threadIdx.x] = acc;
    __syncthreads();
    if (threadIdx.x == 0) {
        int run = partial[blockIdx.x];
        for (int t = 0; t < TPB; ++t) { int tmp = s[t]; s[t] = run; run += tmp; }
    }
    __syncthreads();
    int run = s[threadIdx.x];
    for (int k = tlo; k < thi; ++k) { int tmp = a[k]; a[k] = run; run += tmp; }
}

// ---------------- radix sort: stable multi-round scatter ----------------
// Tile order = (round, wave, lane); async-staged pairs; wave32 ballot ranking.
__global__ void fwm_scatter(const int2* __restrict__ pairIn, int N, int shift, int NB,
                            const int* __restrict__ histScan,
                            int2* __restrict__ pairOut, int* __restrict__ payOnly) {
    __shared__ int2 tileP[TILE];        // 8 KB
    __shared__ int  waveHist[8 * 256];  // 8 KB
    __shared__ int  runBase[256];       // 1 KB
    int tid  = threadIdx.x;
    int base = blockIdx.x * TILE;

    // cp.async-style stage of the whole tile into LDS
    for (int r = 0; r < ITEMS; ++r) {
        int local = r * TPB + tid;
        int i = base + local;
        if (i < N) {
            uint32_t ldsoff = (uint32_t)(uintptr_t)(&tileP[local]);
            const int2* gp = pairIn + i;
            asm volatile("global_load_async_to_lds_b64 %0, %1, off"
                         :: "v"(ldsoff), "v"(gp) : "memory");
        }
    }
    runBase[tid] = 0;
    for (int t = tid; t < 8 * 256; t += TPB) waveHist[t] = 0;
    asm volatile("s_wait_asynccnt 0" ::: "memory");
    __syncthreads();

    int lane = tid & 31;
    int wv   = tid >> 5;
    unsigned ltmask = (1u << lane) - 1u;

    for (int r = 0; r < ITEMS; ++r) {
        int local = r * TPB + tid;
        int i = base + local;
        bool valid = (i < N);
        int2 kp = valid ? tileP[local] : make_int2(0, 0);
        unsigned d = ((unsigned)kp.x >> shift) & 255u;

        unsigned vb = (unsigned)__ballot(valid);
        unsigned mask = 0xFFFFFFFFu;
#pragma unroll
        for (int bit = 0; bit < 8; ++bit) {
            unsigned bb = (unsigned)__ballot(((d >> bit) & 1u) != 0u);
            mask &= ((d >> bit) & 1u) ? bb : ~bb;
        }
        mask &= vb;
        int rank = __popc(mask & ltmask);
        if (valid && rank == 0) waveHist[wv * 256 + (int)d] = __popc(mask);
        __syncthreads();

        if (valid) {
            int below = 0;
            for (int w = 0; w < wv; ++w) below += waveHist[w * 256 + (int)d];
            int pos = histScan[(int)d * NB + blockIdx.x] + runBase[(int)d] + below + rank;
            if (payOnly) payOnly[pos] = kp.y;
            else         pairOut[pos] = kp;
        }
        __syncthreads();

        // fold this round's totals into runBase and clear waveHist for next round
        int tot = 0;
        for (int w = 0; w < 8; ++w) {
            tot += waveHist[w * 256 + tid];
            waveHist[w * 256 + tid] = 0;
        }
        runBase[tid] += tot;
        __syncthreads();
    }
}

// ---------------- host launcher ----------------
extern "C" void kernel_launch(void* const* d_in, const int* in_sizes, int n_in,
                              void* d_out, int out_size, void* d_ws, size_t ws_size,
                              hipStream_t stream) {
    const int* coords = (const int*)d_in[0];
    const int* bszp   = (const int*)d_in[1];
    const int* sshape = (const int*)d_in[2];

    int N  = in_sizes[0] / 4;
    int Np = out_size - 3 * N;
    if (N <= 0 || Np < 0) return;

    int* out = (int*)d_out;
    int* f2w  = out;                 // [Np]
    int* w2f  = out + Np;            // [N]
    int* xmap = out + Np + N;        // [N]
    int* ymap = out + Np + 2 * N;    // [N]

    int* ws = (int*)d_ws;
    int*  meta  = ws;
    int2* pairA = (int2*)(ws + META_INTS);
    int2* pairB = pairA + N;
    int*  hist  = (int*)(pairB + N);
    (void)ws_size;

    int NBt = (N + TILE - 1) / TILE;     // radix tiles
    int NBk = (N + TPB - 1) / TPB;       // keys blocks
    int NBp = (Np + TPB - 1) / TPB;      // flat2win blocks
    int M   = 256 * NBt;                 // hist entries
    int* partial = hist + M;             // [SCB]

    // 1. meta: counts / bs / counts_p / bsp
    hipLaunchKernelGGL(fwm_zero_meta, dim3(1), dim3(META_INTS), 0, stream, meta);
    hipLaunchKernelGGL(fwm_count, dim3(NBt), dim3(TPB), 0, stream, coords, meta, N);
    hipLaunchKernelGGL(fwm_scan_meta, dim3(1), dim3(32), 0, stream, meta, bszp);

    // 2. flat2win over padded range
    if (NBp > 0)
        hipLaunchKernelGGL(fwm_flat2win, dim3(NBp), dim3(TPB), 0, stream, meta, Np, f2w);

    // 3. two argsorts: axis 0 (vx -> xmap), axis 1 (vy -> ymap)
    for (int axis = 0; axis < 2; ++axis) {
        hipLaunchKernelGGL(fwm_keys, dim3(NBk), dim3(TPB), 0, stream,
                           coords, meta, sshape, N, pairA,
                           (axis == 0) ? w2f : (int*)nullptr, axis);
        int* finalPay = (axis == 0) ? xmap : ymap;
        for (int pass = 0; pass < 4; ++pass) {
            int shift = 8 * pass;
            int2* pIn  = (pass & 1) ? pairB : pairA;
            int2* pOut = (pass & 1) ? pairA : pairB;
            hipLaunchKernelGGL(fwm_hist,  dim3(NBt), dim3(TPB), 0, stream, pIn, N, shift, NBt, hist);
            hipLaunchKernelGGL(fwm_scan1, dim3(SCB), dim3(TPB), 0, stream, hist, M, partial);
            hipLaunchKernelGGL(fwm_scan2, dim3(1),   dim3(32),  0, stream, partial);
            hipLaunchKernelGGL(fwm_scan3, dim3(SCB), dim3(TPB), 0, stream, hist, M, partial);
            hipLaunchKernelGGL(fwm_scatter, dim3(NBt), dim3(TPB), 0, stream,
                               pIn, N, shift, NBt, hist, pOut,
                               (pass == 3) ? finalPay : (int*)nullptr);
        }
    }
}